// MyConv1D_64304250356384
// MI455X (gfx1250) — compile-verified
//
#include <hip/hip_runtime.h>

// ---------------------------------------------------------------------------
// Conv1D (B=32, SEQ=16384, C_IN=64, F=64, K=9, VALID) as implicit GEMM on the
// CDNA5 f32 WMMA pipe:  out[b,t,f] = sum_{k,c} x[b,t+k,c] * w[f,k,c] + bias[f]
//   M = output positions, N = 64 filters, K' = 9*64 = 576.
// Register blocking: each wave owns a 32(M) x 64(N) tile = 8x v8f accumulators
// (two 16-row subtiles reuse every B fragment -> halves LDS bytes per WMMA).
// Weights staged in LDS with a 580-dword row pitch: 580 % 64 = 4, so the 16
// B-fragment lanes (stride = pitch) land on banks {4n,4n+1} and lanes 16-31
// on {4n+2,4n+3} -> all 64 banks distinct, conflict-free ds_load_b64.
// ---------------------------------------------------------------------------

typedef __attribute__((ext_vector_type(2))) float v2f;
typedef __attribute__((ext_vector_type(8))) float v8f;

#define CONV_B     32
#define CONV_SEQ   16384
#define CONV_C     64
#define CONV_F     64
#define CONV_K     9
#define CONV_TOUT  (CONV_SEQ - CONV_K + 1)   // 16376
#define CONV_KC    (CONV_K * CONV_C)         // 576
#define LDS_PITCH  580                       // 576 padded: bank-conflict-free

__global__ __launch_bounds__(128) void conv1d_wmma_f32(
    const float* __restrict__ x,      // [B, SEQ, C]
    const float* __restrict__ w,      // [F, K, C] (row = 576 contiguous f32)
    const float* __restrict__ bias,   // [F]
    float* __restrict__ out)          // [B, TOUT, F]
{
    __shared__ float wlds[CONV_F * LDS_PITCH];   // 148,480 B of 320 KB WGP LDS

    const int tid = threadIdx.x;

    // ---- cooperative weight stage: 36864 floats, float4 granularity -------
    // 576 % 4 == 0, so a float4 never crosses a filter row; LDS offset
    // f*580 + kk is a multiple of 4 dwords -> 16B-aligned ds_store_b128.
    for (int i = tid * 4; i < CONV_F * CONV_KC; i += 128 * 4) {
        const float4 v = *reinterpret_cast<const float4*>(w + i);
        const int f  = i / CONV_KC;
        const int kk = i - f * CONV_KC;
        *reinterpret_cast<float4*>(&wlds[f * LDS_PITCH + kk]) = v;
    }
    __syncthreads();

    // ---- per-wave tile: M=32 (two 16-row subtiles), N=64 (four 16-col) ----
    const int lane = tid & 31;
    const int wave = tid >> 5;           // 4 waves per block
    const int m    = lane & 15;          // A: M index / B: N index
    const int half = lane >> 4;          // K-pair select (ISA A/B layout)

    const int b  = blockIdx.y;                       // batch
    const int t0 = (blockIdx.x * 4 + wave) * 32;     // first output position

    const float* xb = x + (size_t)b * CONV_SEQ * CONV_C;

    // Accumulators pre-loaded with bias: element (r, lane) of tile (s,j) is
    // filter f = j*16 + (lane&15), identical across all 8 VGPRs.
    v8f acc[2][4];
#pragma unroll
    for (int j = 0; j < 4; ++j) {
        const float bv = bias[j * 16 + m];
        v8f a;
#pragma unroll
        for (int r = 0; r < 8; ++r) a[r] = bv;
        acc[0][j] = a;
        acc[1][j] = a;
    }

    // Per-lane LDS base (dwords): B[kk, n] fragment element for filter group j
    // lives at (j*16 + m)*580 + kk + 2*half  (all terms even -> 8B aligned).
    const int bbase = m * LDS_PITCH + 2 * half;

#pragma unroll 1
    for (int k = 0; k < CONV_K; ++k) {
        // Clamp keeps the tail tile's loads inside this batch's x; the junk
        // rows it produces are masked at store time (t >= 16376 never stored).
        int row0 = t0 + m + k;
        int row1 = t0 + 16 + m + k;
        if (row0 > CONV_SEQ - 1) row0 = CONV_SEQ - 1;
        if (row1 > CONV_SEQ - 1) row1 = CONV_SEQ - 1;
        const float* pa0 = xb + (size_t)row0 * CONV_C + 2 * half;
        const float* pa1 = xb + (size_t)row1 * CONV_C + 2 * half;
        const int    bk  = bbase + k * CONV_C;

#pragma unroll
        for (int c0 = 0; c0 < CONV_C; c0 += 4) {
            // A fragments: 16x4 f32 slices of x (lane l: M=l%16, K=2*(l/16)+v)
            const v2f a0 = *reinterpret_cast<const v2f*>(pa0 + c0);
            const v2f a1 = *reinterpret_cast<const v2f*>(pa1 + c0);
#pragma unroll
            for (int j = 0; j < 4; ++j) {
                // B fragment: 4x16 f32 slice of w (lane l: N=l%16, K=2*(l/16)+v)
                const v2f bf = *reinterpret_cast<const v2f*>(
                    &wlds[bk + c0 + j * 16 * LDS_PITCH]);
                acc[0][j] = __builtin_amdgcn_wmma_f32_16x16x4_f32(
                    false, a0, false, bf, (short)0, acc[0][j], false, false);
                acc[1][j] = __builtin_amdgcn_wmma_f32_16x16x4_f32(
                    false, a1, false, bf, (short)0, acc[1][j], false, false);
            }
        }
    }

    // ---- store: C/D layout -> lane l, vgpr r = (M = r + 8*(l/16), N = l%16)
    float* ob = out + (size_t)b * CONV_TOUT * CONV_F;
#pragma unroll
    for (int s = 0; s < 2; ++s) {
#pragma unroll
        for (int r = 0; r < 8; ++r) {
            const int t = t0 + 16 * s + r + 8 * half;
            if (t < CONV_TOUT) {
#pragma unroll
                for (int j = 0; j < 4; ++j) {
                    ob[(size_t)t * CONV_F + j * 16 + m] = acc[s][j][r];
                }
            }
        }
    }
}

extern "C" void kernel_launch(void* const* d_in, const int* in_sizes, int n_in,
                              void* d_out, int out_size, void* d_ws, size_t ws_size,
                              hipStream_t stream) {
    const float* x    = (const float*)d_in[0];   // [32,16384,64]
    const float* w    = (const float*)d_in[1];   // [64,9,64]
    const float* bias = (const float*)d_in[2];   // [64]
    float*       out  = (float*)d_out;           // [32,16376,64]

    // 128 M-tiles of 128 positions per batch (128*128 = 16384 >= 16376), 32 batches.
    dim3 grid(CONV_SEQ / 128, CONV_B);
    conv1d_wmma_f32<<<grid, 128, 0, stream>>>(x, w, bias, out);
}